// RecGNN_63977832841246
// MI455X (gfx1250) — compile-verified
//
#include <hip/hip_runtime.h>

// ---------------------------------------------------------------------------
// EGAT x2 + segment pool + batchnorm for MI455X (gfx1250, wave32).
// GEMMs via V_WMMA_F32_16X16X4_F32 (fp32 in/out, matches fp32 reference).
// Inner GEMM loop is branch-free: clamped unconditional loads (b64 for A),
// K-remainder handled in a single wave-uniform tail with v_cndmask selects.
// Edge softmax / message passing via wave-per-edge kernels with b128 loads,
// butterfly wave reductions, and global float atomics.
// ---------------------------------------------------------------------------

typedef float v2f __attribute__((ext_vector_type(2)));
typedef float v8f __attribute__((ext_vector_type(8)));

#define SLOPE 0.01f

__device__ __forceinline__ float leaky(float x) { return x > 0.f ? x : SLOPE * x; }

__device__ __forceinline__ float waveSum(float v) {
  v += __shfl_xor(v, 16, 32);
  v += __shfl_xor(v, 8, 32);
  v += __shfl_xor(v, 4, 32);
  v += __shfl_xor(v, 2, 32);
  v += __shfl_xor(v, 1, 32);
  return v;
}

// monotonic float -> uint key so atomicMax(u32) implements float max;
// key 0 decodes below every real float, so memset(0) is the -inf init.
__device__ __forceinline__ unsigned f2key(float f) {
  unsigned u = __float_as_uint(f);
  return (u & 0x80000000u) ? ~u : (u | 0x80000000u);
}
__device__ __forceinline__ float key2f(unsigned k) {
  return (k & 0x80000000u) ? __uint_as_float(k ^ 0x80000000u) : __uint_as_float(~k);
}

// ---------------------------------------------------------------------------
// C[M,N] = A[M,K] @ B[K,N], row-major, one 16x16 C tile per wave.
// 4 waves per block cover 4 consecutive N-tiles (blockIdx.y strip).
// Lane layout (wave32): lo = lane&15, hi = lane>>4.
//   A operand: lane holds {A[m][k0+2*hi], A[m][k0+2*hi+1]}   (one b64 load)
//   B operand: lane holds {B[k0+2*hi][n], B[k0+2*hi+1][n]}   (two b32 loads)
//   C VGPR j -> C[tile_m + j + 8*hi][n]
// ---------------------------------------------------------------------------
__global__ __launch_bounds__(128) void gemm_wmma_f32(
    const float* __restrict__ A, const float* __restrict__ B,
    float* __restrict__ C, int M, int K, int N) {
  const int wave = threadIdx.x >> 5;
  const int lane = threadIdx.x & 31;
  const int tn = blockIdx.y * 4 + wave;
  const int tm = blockIdx.x;
  if (tn * 16 >= N) return;              // wave-uniform exit, EXEC stays all-1s
  const int hi = lane >> 4;
  const int lo = lane & 15;
  const int m = tm * 16 + lo;
  const int n = tn * 16 + lo;
  // clamp A row: always-in-bounds loads; clamped rows are never stored.
  const int mc = (m < M) ? m : (M - 1);
  const float* __restrict__ Arow = A + (size_t)mc * K;
  const float* __restrict__ Bcol = B + n;

  const int Kmain = K & ~3;
  v8f acc = {0.f, 0.f, 0.f, 0.f, 0.f, 0.f, 0.f, 0.f};

  int k0 = 0;
  // 2x unrolled branch-free main loop (A: b64 load, B: 2x b32 loads, 1 wmma each)
  for (; k0 + 8 <= Kmain; k0 += 8) {
    const int ka0 = k0 + hi * 2;
    const int ka1 = k0 + 4 + hi * 2;
    const v2f a0 = *reinterpret_cast<const v2f*>(Arow + ka0);
    const v2f a1 = *reinterpret_cast<const v2f*>(Arow + ka1);
    v2f b0, b1;
    b0.x = Bcol[(size_t)ka0 * N];
    b0.y = Bcol[(size_t)(ka0 + 1) * N];
    b1.x = Bcol[(size_t)ka1 * N];
    b1.y = Bcol[(size_t)(ka1 + 1) * N];
    acc = __builtin_amdgcn_wmma_f32_16x16x4_f32(
        false, a0, false, b0, (short)0, acc, false, false);
    acc = __builtin_amdgcn_wmma_f32_16x16x4_f32(
        false, a1, false, b1, (short)0, acc, false, false);
  }
  for (; k0 < Kmain; k0 += 4) {
    const int ka = k0 + hi * 2;
    const v2f a = *reinterpret_cast<const v2f*>(Arow + ka);
    v2f b;
    b.x = Bcol[(size_t)ka * N];
    b.y = Bcol[(size_t)(ka + 1) * N];
    acc = __builtin_amdgcn_wmma_f32_16x16x4_f32(
        false, a, false, b, (short)0, acc, false, false);
  }
  if (Kmain < K) {                       // wave-uniform K tail (only K=34 here)
    const int ka = Kmain + hi * 2;
    const int i0 = (ka < K) ? ka : (K - 1);          // clamped, in-bounds
    const int i1 = (ka + 1 < K) ? (ka + 1) : (K - 1);
    const float ax = Arow[i0], ay = Arow[i1];        // unconditional loads
    const float bx = Bcol[(size_t)i0 * N], by = Bcol[(size_t)i1 * N];
    v2f a, b;
    a.x = (ka < K) ? ax : 0.f;                       // pure v_cndmask selects
    a.y = (ka + 1 < K) ? ay : 0.f;
    b.x = (ka < K) ? bx : 0.f;
    b.y = (ka + 1 < K) ? by : 0.f;
    acc = __builtin_amdgcn_wmma_f32_16x16x4_f32(
        false, a, false, b, (short)0, acc, false, false);
  }

  float* __restrict__ Crow = C + (size_t)(tm * 16 + hi * 8) * N + n;
  if ((tm + 1) * 16 <= M) {              // wave-uniform: full M tile (common)
#pragma unroll
    for (int j = 0; j < 8; ++j) Crow[(size_t)j * N] = acc[j];
  } else {                               // last partial tile only
#pragma unroll
    for (int j = 0; j < 8; ++j)
      if (tm * 16 + hi * 8 + j < M) Crow[(size_t)j * N] = acc[j];
  }
}

// ---------------------------------------------------------------------------
// Layer 1 edge kernel: f1[e,c] = leaky(fni[src,c] + fnj[dst,c] + (ef@Wf1)[c]
//                                      + bias[c]); scores + segment max.
// One wave per edge; lane l covers channels l (head0) and 32+l (head1).
// ---------------------------------------------------------------------------
__global__ __launch_bounds__(256) void edge1_kernel(
    const float* __restrict__ fni, const float* __restrict__ fnj,
    const float* __restrict__ efeat, const float* __restrict__ Wf,
    const float* __restrict__ bias, const float* __restrict__ attn,
    const int* __restrict__ src, const int* __restrict__ dst,
    float* __restrict__ f1, float* __restrict__ score,
    unsigned* __restrict__ nmax, int E) {
  const int e = blockIdx.x * 8 + (threadIdx.x >> 5);
  if (e >= E) return;
  const int l = threadIdx.x & 31;
  const int s = src[e], d = dst[e];
  float ef[7];
#pragma unroll
  for (int j = 0; j < 7; ++j) ef[j] = efeat[(size_t)e * 7 + j];
  float sc[2];
#pragma unroll
  for (int h = 0; h < 2; ++h) {
    const int c = h * 32 + l;
    float f = fni[(size_t)s * 64 + c] + fnj[(size_t)d * 64 + c] + bias[c];
#pragma unroll
    for (int j = 0; j < 7; ++j) f += ef[j] * Wf[j * 64 + c];
    f = leaky(f);
    f1[(size_t)e * 64 + c] = f;
    sc[h] = f * attn[c];                 // attn1 flat (2,32) == channel index
  }
  const float s0 = waveSum(sc[0]);
  const float s1 = waveSum(sc[1]);
  if (l == 0) {
    score[(size_t)e * 2 + 0] = s0;
    score[(size_t)e * 2 + 1] = s1;
    atomicMax(&nmax[(size_t)d * 2 + 0], f2key(s0));
    atomicMax(&nmax[(size_t)d * 2 + 1], f2key(s1));
  }
}

// ---------------------------------------------------------------------------
// ex = exp(score - segmax[dst]); in-place into score; segment-sum into nsum.
// ---------------------------------------------------------------------------
__global__ __launch_bounds__(256) void expsum_kernel(
    float* __restrict__ score, const unsigned* __restrict__ nmax,
    const int* __restrict__ dst, float* __restrict__ nsum, int E) {
  const int i = blockIdx.x * 256 + threadIdx.x;
  if (i >= 2 * E) return;
  const int e = i >> 1, h = i & 1;
  const int d = dst[e];
  const float m = key2f(nmax[(size_t)d * 2 + h]);
  const float ex = expf(score[i] - m);
  score[i] = ex;
  atomicAdd(&nsum[(size_t)d * 2 + h], ex);
}

// ---------------------------------------------------------------------------
// Layer 1 message passing: h1[dst, h*128+c] += hsrc[src, h*128+c] * a[e,h].
// One wave per edge; two float4 (b128) loads per lane, 8 atomic f32 adds.
// ---------------------------------------------------------------------------
__global__ __launch_bounds__(256) void message1_kernel(
    const float* __restrict__ hsrc, const float* __restrict__ exsc,
    const float* __restrict__ nsum, const int* __restrict__ src,
    const int* __restrict__ dst, float* __restrict__ h1, int E) {
  const int e = blockIdx.x * 8 + (threadIdx.x >> 5);
  if (e >= E) return;
  const int l = threadIdx.x & 31;
  const int s = src[e], d = dst[e];
  const float a0 = exsc[(size_t)e * 2 + 0] / nsum[(size_t)d * 2 + 0];
  const float a1 = exsc[(size_t)e * 2 + 1] / nsum[(size_t)d * 2 + 1];
  const float4* hs = reinterpret_cast<const float4*>(hsrc + (size_t)s * 256);
  float* out = h1 + (size_t)d * 256;
  const float4 x0 = hs[l];               // head0: c = l*4
  const float4 x1 = hs[32 + l];          // head1: c = 128 + l*4
  atomicAdd(out + l * 4 + 0, x0.x * a0);
  atomicAdd(out + l * 4 + 1, x0.y * a0);
  atomicAdd(out + l * 4 + 2, x0.z * a0);
  atomicAdd(out + l * 4 + 3, x0.w * a0);
  atomicAdd(out + 128 + l * 4 + 0, x1.x * a1);
  atomicAdd(out + 128 + l * 4 + 1, x1.y * a1);
  atomicAdd(out + 128 + l * 4 + 2, x1.z * a1);
  atomicAdd(out + 128 + l * 4 + 3, x1.w * a1);
}

// ---------------------------------------------------------------------------
// Layer 2 edge kernel: scores from leaky(fni2[src]+fnj2[dst]+fe2[e]+bias2).
// (e2 itself is dead in the reference; only scores survive.)
// ---------------------------------------------------------------------------
__global__ __launch_bounds__(256) void edge2_kernel(
    const float* __restrict__ fni, const float* __restrict__ fnj,
    const float* __restrict__ fe, const float* __restrict__ bias,
    const float* __restrict__ attn, const int* __restrict__ src,
    const int* __restrict__ dst, float* __restrict__ score,
    unsigned* __restrict__ nmax, int E) {
  const int e = blockIdx.x * 8 + (threadIdx.x >> 5);
  if (e >= E) return;
  const int l = threadIdx.x & 31;
  const int s = src[e], d = dst[e];
  const float4* pi = reinterpret_cast<const float4*>(fni + (size_t)s * 256);
  const float4* pj = reinterpret_cast<const float4*>(fnj + (size_t)d * 256);
  const float4* pe = reinterpret_cast<const float4*>(fe + (size_t)e * 256);
  const float4* pb = reinterpret_cast<const float4*>(bias);
  const float4* pa = reinterpret_cast<const float4*>(attn);
  float sh[2];
#pragma unroll
  for (int h = 0; h < 2; ++h) {
    const int idx = h * 32 + l;          // float4 index; c = idx*4
    const float4 fi = pi[idx], fj = pj[idx], fx = pe[idx];
    const float4 fb = pb[idx], fa = pa[idx];
    const float f0 = leaky(fi.x + fj.x + fx.x + fb.x);
    const float f1 = leaky(fi.y + fj.y + fx.y + fb.y);
    const float f2 = leaky(fi.z + fj.z + fx.z + fb.z);
    const float f3 = leaky(fi.w + fj.w + fx.w + fb.w);
    sh[h] = f0 * fa.x + f1 * fa.y + f2 * fa.z + f3 * fa.w;
  }
  const float s0 = waveSum(sh[0]);
  const float s1 = waveSum(sh[1]);
  if (l == 0) {
    score[(size_t)e * 2 + 0] = s0;
    score[(size_t)e * 2 + 1] = s1;
    atomicMax(&nmax[(size_t)d * 2 + 0], f2key(s0));
    atomicMax(&nmax[(size_t)d * 2 + 1], f2key(s1));
  }
}

// ---------------------------------------------------------------------------
// Layer 2 message passing with fused head-mean:
//   h2mean[dst,c] += 0.5*(a0*hsrc2[src,c] + a1*hsrc2[src,512+c]), c<512.
// Also emits the normalized attention (second tuple output).
// ---------------------------------------------------------------------------
__global__ __launch_bounds__(256) void message2_kernel(
    const float* __restrict__ hsrc, const float* __restrict__ exsc,
    const float* __restrict__ nsum, const int* __restrict__ src,
    const int* __restrict__ dst, float* __restrict__ h2mean,
    float* __restrict__ attn_out, int E) {
  const int e = blockIdx.x * 8 + (threadIdx.x >> 5);
  if (e >= E) return;
  const int l = threadIdx.x & 31;
  const int s = src[e], d = dst[e];
  const float a0 = exsc[(size_t)e * 2 + 0] / nsum[(size_t)d * 2 + 0];
  const float a1 = exsc[(size_t)e * 2 + 1] / nsum[(size_t)d * 2 + 1];
  if (l == 0) {
    attn_out[(size_t)e * 2 + 0] = a0;
    attn_out[(size_t)e * 2 + 1] = a1;
  }
  const float4* hs = reinterpret_cast<const float4*>(hsrc + (size_t)s * 1024);
  float* out = h2mean + (size_t)d * 512;
#pragma unroll
  for (int q = 0; q < 4; ++q) {
    const int ci = q * 32 + l;           // float4 index within 512 channels
    const float4 x0 = hs[ci];            // head0
    const float4 x1 = hs[128 + ci];      // head1
    atomicAdd(out + ci * 4 + 0, 0.5f * (a0 * x0.x + a1 * x1.x));
    atomicAdd(out + ci * 4 + 1, 0.5f * (a0 * x0.y + a1 * x1.y));
    atomicAdd(out + ci * 4 + 2, 0.5f * (a0 * x0.z + a1 * x1.z));
    atomicAdd(out + ci * 4 + 3, 0.5f * (a0 * x0.w + a1 * x1.w));
  }
}

__global__ void prefix_kernel(const int* __restrict__ bnn, int* __restrict__ offs,
                              int Bc) {
  if (threadIdx.x == 0 && blockIdx.x == 0) {
    int sum = 0;
    for (int b = 0; b < Bc; ++b) { offs[b] = sum; sum += bnn[b]; }
    offs[Bc] = sum;
  }
}

// pooled[b,c] = sum over nodes of graph b of h2mean[n,c]
__global__ __launch_bounds__(256) void pool_kernel(
    const float* __restrict__ h2, const int* __restrict__ offs,
    float* __restrict__ pooled) {
  const int b = blockIdx.x;
  const int c = blockIdx.y * 256 + threadIdx.x;
  const int n0 = offs[b], n1 = offs[b + 1];
  float s = 0.f;
  for (int n = n0; n < n1; ++n) s += h2[(size_t)n * 512 + c];
  pooled[(size_t)b * 512 + c] = s;
}

// BatchNorm1d over batch dim (training-mode batch stats, biased variance)
__global__ __launch_bounds__(512) void bn_kernel(
    const float* __restrict__ pooled, const float* __restrict__ gamma,
    const float* __restrict__ beta, float* __restrict__ out, int Bc) {
  const int c = threadIdx.x;
  float mu = 0.f;
  for (int b = 0; b < Bc; ++b) mu += pooled[(size_t)b * 512 + c];
  mu /= (float)Bc;
  float var = 0.f;
  for (int b = 0; b < Bc; ++b) {
    const float t = pooled[(size_t)b * 512 + c] - mu;
    var += t * t;
  }
  var /= (float)Bc;
  const float inv = rsqrtf(var + 1e-5f);
  const float g = gamma[c], be = beta[c];
  for (int b = 0; b < Bc; ++b)
    out[(size_t)b * 512 + c] = (pooled[(size_t)b * 512 + c] - mu) * inv * g + be;
}

extern "C" void kernel_launch(void* const* d_in, const int* in_sizes, int n_in,
                              void* d_out, int out_size, void* d_ws, size_t ws_size,
                              hipStream_t stream) {
  const float* n_feats = (const float*)d_in[0];
  const float* e_feats = (const float*)d_in[1];
  const int* src       = (const int*)d_in[2];
  const int* dst       = (const int*)d_in[3];
  const int* bnn       = (const int*)d_in[4];
  const float* Wni1    = (const float*)d_in[5];
  const float* Wnj1    = (const float*)d_in[6];
  const float* Wf1     = (const float*)d_in[7];
  const float* Wsrc1   = (const float*)d_in[8];
  const float* attn1   = (const float*)d_in[9];
  const float* bias1   = (const float*)d_in[10];
  const float* Wni2    = (const float*)d_in[11];
  const float* Wnj2    = (const float*)d_in[12];
  const float* Wf2     = (const float*)d_in[13];
  const float* Wsrc2   = (const float*)d_in[14];
  const float* attn2   = (const float*)d_in[15];
  const float* bias2   = (const float*)d_in[16];
  const float* gamma   = (const float*)d_in[17];
  const float* beta    = (const float*)d_in[18];

  const int E  = in_sizes[2];        // 400000
  const int Nn = in_sizes[0] / 34;   // 50000
  const int Bc = in_sizes[4];        // 50

  size_t off = 0;
  auto carve = [&](size_t bytes) -> void* {
    void* p = (void*)((char*)d_ws + off);
    off += (bytes + 255) & ~(size_t)255;
    return p;
  };
  float* fni1     = (float*)carve((size_t)Nn * 64 * 4);
  float* fnj1     = (float*)carve((size_t)Nn * 64 * 4);
  float* hsrc1    = (float*)carve((size_t)Nn * 256 * 4);
  float* f1       = (float*)carve((size_t)E * 64 * 4);     // e1 for layer 2
  float* score1   = (float*)carve((size_t)E * 2 * 4);
  unsigned* nmax1 = (unsigned*)carve((size_t)Nn * 2 * 4);
  float* nsum1    = (float*)carve((size_t)Nn * 2 * 4);
  float* h1       = (float*)carve((size_t)Nn * 256 * 4);
  float* fni2     = (float*)carve((size_t)Nn * 256 * 4);
  float* fnj2     = (float*)carve((size_t)Nn * 256 * 4);
  float* fe2      = (float*)carve((size_t)E * 256 * 4);
  float* hsrc2    = (float*)carve((size_t)Nn * 1024 * 4);
  float* score2   = (float*)carve((size_t)E * 2 * 4);
  unsigned* nmax2 = (unsigned*)carve((size_t)Nn * 2 * 4);
  float* nsum2    = (float*)carve((size_t)Nn * 2 * 4);
  float* h2m      = (float*)carve((size_t)Nn * 512 * 4);
  int* offs       = (int*)carve((size_t)(Bc + 1) * 4);
  float* pooled   = (float*)carve((size_t)Bc * 512 * 4);

  float* out_bn   = (float*)d_out;                    // (B, 512)
  float* attn_out = (float*)d_out + (size_t)Bc * 512; // (E, 2, 1)

  hipMemsetAsync(nmax1, 0, (size_t)Nn * 2 * 4, stream);  // key 0 == -inf
  hipMemsetAsync(nsum1, 0, (size_t)Nn * 2 * 4, stream);
  hipMemsetAsync(nmax2, 0, (size_t)Nn * 2 * 4, stream);
  hipMemsetAsync(nsum2, 0, (size_t)Nn * 2 * 4, stream);
  hipMemsetAsync(h1,    0, (size_t)Nn * 256 * 4, stream);
  hipMemsetAsync(h2m,   0, (size_t)Nn * 512 * 4, stream);

  const dim3 blk128(128), blk256(256);
  const int mt_n = (Nn + 15) / 16;
  const int mt_e = (E + 15) / 16;
  const int eg   = (E + 7) / 8;

  // -------- layer 1 dense (WMMA) --------
  gemm_wmma_f32<<<dim3(mt_n, 1),  blk128, 0, stream>>>(n_feats, Wni1,  fni1,  Nn, 34, 64);
  gemm_wmma_f32<<<dim3(mt_n, 1),  blk128, 0, stream>>>(n_feats, Wnj1,  fnj1,  Nn, 34, 64);
  gemm_wmma_f32<<<dim3(mt_n, 4),  blk128, 0, stream>>>(n_feats, Wsrc1, hsrc1, Nn, 34, 256);

  // -------- layer 1 edge softmax + aggregate --------
  edge1_kernel<<<eg, blk256, 0, stream>>>(fni1, fnj1, e_feats, Wf1, bias1, attn1,
                                          src, dst, f1, score1, nmax1, E);
  expsum_kernel<<<(2 * E + 255) / 256, blk256, 0, stream>>>(score1, nmax1, dst, nsum1, E);
  message1_kernel<<<eg, blk256, 0, stream>>>(hsrc1, score1, nsum1, src, dst, h1, E);

  // -------- layer 2 dense (WMMA) --------
  gemm_wmma_f32<<<dim3(mt_n, 4),  blk128, 0, stream>>>(h1, Wni2,  fni2,  Nn, 256, 256);
  gemm_wmma_f32<<<dim3(mt_n, 4),  blk128, 0, stream>>>(h1, Wnj2,  fnj2,  Nn, 256, 256);
  gemm_wmma_f32<<<dim3(mt_e, 4),  blk128, 0, stream>>>(f1, Wf2,   fe2,   E,  64,  256);
  gemm_wmma_f32<<<dim3(mt_n, 16), blk128, 0, stream>>>(h1, Wsrc2, hsrc2, Nn, 256, 1024);

  // -------- layer 2 edge softmax + aggregate (fused head mean) --------
  edge2_kernel<<<eg, blk256, 0, stream>>>(fni2, fnj2, fe2, bias2, attn2, src, dst,
                                          score2, nmax2, E);
  expsum_kernel<<<(2 * E + 255) / 256, blk256, 0, stream>>>(score2, nmax2, dst, nsum2, E);
  message2_kernel<<<eg, blk256, 0, stream>>>(hsrc2, score2, nsum2, src, dst, h2m,
                                             attn_out, E);

  // -------- pool per graph + batchnorm --------
  prefix_kernel<<<1, 1, 0, stream>>>(bnn, offs, Bc);
  pool_kernel<<<dim3(Bc, 2), blk256, 0, stream>>>(h2m, offs, pooled);
  bn_kernel<<<1, 512, 0, stream>>>(pooled, gamma, beta, out_bn, Bc);
}